// GCNFraudDetector_34660386078849
// MI455X (gfx1250) — compile-verified
//
#include <hip/hip_runtime.h>
#include <hip/hip_bf16.h>

typedef float v2f __attribute__((ext_vector_type(2)));
typedef float v8f __attribute__((ext_vector_type(8)));

#define N_CH0 128
#define EPS_BN 1e-5f

// ---------------------------------------------------------------------------
// Degree / normalization
// ---------------------------------------------------------------------------
__global__ void init_deg_kernel(float* deg, int n) {
    int i = blockIdx.x * blockDim.x + threadIdx.x;
    if (i < n) deg[i] = 1.0f;  // self loop
}

__global__ void deg_scatter_kernel(const long long* __restrict__ ei, float* deg, int nedges) {
    int e = blockIdx.x * blockDim.x + threadIdx.x;
    if (e < nedges) {
        int d = (int)ei[(size_t)nedges + e];  // dst row of edge_index (2,E)
        atomicAdd(&deg[d], 1.0f);
    }
}

__global__ void finalize_dis_kernel(float* dis, int n) {
    int i = blockIdx.x * blockDim.x + threadIdx.x;
    if (i < n) dis[i] = rsqrtf(fmaxf(dis[i], 1e-12f));
}

// ---------------------------------------------------------------------------
// Async copy of 16B global -> LDS (CDNA5, tracked with ASYNCcnt)
// ---------------------------------------------------------------------------
__device__ __forceinline__ void async_copy_b128(uint32_t lds_byte_addr, const float* gsrc) {
    asm volatile("global_load_async_to_lds_b128 %0, %1, off"
                 :
                 : "v"(lds_byte_addr), "v"(gsrc)
                 : "memory");
}

__device__ __forceinline__ void wait_asynccnt0() {
#if __has_builtin(__builtin_amdgcn_s_wait_asynccnt)
    __builtin_amdgcn_s_wait_asynccnt(0);
#else
    asm volatile("s_wait_asynccnt 0x0" ::: "memory");
#endif
}

// ---------------------------------------------------------------------------
// WMMA fp32 GEMM: C[nrows x NOUT] = A[nrows x K] @ W[K x NOUT]
// Block = MTILES x 16 rows; wave w covers cols [16w, 16w+16) and keeps MTILES
// accumulators so each B fragment feeds MTILES WMMAs. A tile staged in LDS via
// async-to-LDS (fast path) or guarded scalar loads (tail block).
// ---------------------------------------------------------------------------
template <int K, int NOUT, int MTILES>
__global__ __launch_bounds__(32 * (NOUT / 16)) void gemm_wmma_kernel(
    const float* __restrict__ A, const float* __restrict__ W,
    float* __restrict__ C, int nrows) {
    constexpr int WAVES = NOUT / 16;
    constexpr int NTHR = 32 * WAVES;
    constexpr int ROWS = 16 * MTILES;
    __shared__ float As[ROWS * K];

    const int row0 = blockIdx.x * ROWS;
    const int tid = threadIdx.x;
    const bool full_tile = (row0 + ROWS) <= nrows;

    if (full_tile) {
        // Rows of this tile are contiguous in memory: bulk async 16B copies.
        constexpr int NV4 = ROWS * K / 4;
        const uint32_t lds_base = (uint32_t)(uintptr_t)As;
        const float* gbase = A + (size_t)row0 * K;
#pragma unroll
        for (int it = 0; it < NV4 / NTHR; ++it) {
            const int idx4 = tid + it * NTHR;
            async_copy_b128(lds_base + (uint32_t)idx4 * 16u, gbase + (size_t)idx4 * 4);
        }
        wait_asynccnt0();
    } else {
        for (int idx = tid; idx < ROWS * K; idx += NTHR) {
            int r = idx / K, k = idx - r * K;
            int gr = row0 + r;
            As[idx] = (gr < nrows) ? A[(size_t)gr * K + k] : 0.0f;
        }
    }
    __syncthreads();

    const int wave = tid >> 5;
    const int lane = tid & 31;
    const int half = lane >> 4;  // 0: lanes 0-15, 1: lanes 16-31
    const int lr = lane & 15;
    const int col = wave * 16 + lr;

    v8f acc[MTILES];
#pragma unroll
    for (int m = 0; m < MTILES; ++m) acc[m] = (v8f){};

#pragma unroll 2
    for (int k0 = 0; k0 < K; k0 += 4) {
        const int ka = k0 + 2 * half;
        v2f b;
        b[0] = W[(size_t)ka * NOUT + col];
        b[1] = W[(size_t)(ka + 1) * NOUT + col];
#pragma unroll
        for (int m = 0; m < MTILES; ++m) {
            const float2 av = *(const float2*)&As[(m * 16 + lr) * K + ka];
            v2f a;
            a[0] = av.x;
            a[1] = av.y;
            acc[m] = __builtin_amdgcn_wmma_f32_16x16x4_f32(
                /*neg_a=*/false, a, /*neg_b=*/false, b,
                /*c_mod=*/(short)0, acc[m], /*reuse_a=*/false, /*reuse_b=*/false);
        }
    }

    if (full_tile) {
#pragma unroll
        for (int m = 0; m < MTILES; ++m)
#pragma unroll
            for (int v = 0; v < 8; ++v)
                C[(size_t)(row0 + m * 16 + v + 8 * half) * NOUT + col] = acc[m][v];
    } else {
#pragma unroll
        for (int m = 0; m < MTILES; ++m)
#pragma unroll
            for (int v = 0; v < 8; ++v) {
                int r = row0 + m * 16 + v + 8 * half;
                if (r < nrows) C[(size_t)r * NOUT + col] = acc[m][v];
            }
    }
}

// ---------------------------------------------------------------------------
// agg[i,c] = t[i,c] * dis[i]^2 + bias[c]   (self-loop + bias fused)
// ---------------------------------------------------------------------------
template <int NC>
__global__ void agg_init_kernel(const float* __restrict__ t, const float* __restrict__ dis,
                                const float* __restrict__ bias, float* __restrict__ agg, int n) {
    size_t gid = (size_t)blockIdx.x * blockDim.x + threadIdx.x;
    if (gid >= (size_t)n * NC) return;
    int i = (int)(gid / NC);
    int c = (int)(gid - (size_t)i * NC);
    float di = dis[i];
    agg[gid] = t[gid] * di * di + bias[c];
}

// ---------------------------------------------------------------------------
// Edge scatter: agg[dst] += t[src] * dis[src]*dis[dst]; float4 per thread.
// ---------------------------------------------------------------------------
template <int NC>
__global__ void edge_scatter_kernel(const long long* __restrict__ ei,
                                    const float* __restrict__ t, const float* __restrict__ dis,
                                    float* __restrict__ agg, int nedges) {
    constexpr int G = NC / 4;
    long long gid = (long long)blockIdx.x * blockDim.x + threadIdx.x;
    if (gid >= (long long)nedges * G) return;
    int e = (int)(gid / G);
    int c4 = (int)(gid - (long long)e * G);
    int s = (int)ei[e];
    int d = (int)ei[(size_t)nedges + e];
    float nrm = dis[s] * dis[d];
    const float4 v = *(const float4*)(t + (size_t)s * NC + c4 * 4);
    float* o = agg + (size_t)d * NC + c4 * 4;
    atomicAdd(o + 0, v.x * nrm);
    atomicAdd(o + 1, v.y * nrm);
    atomicAdd(o + 2, v.z * nrm);
    atomicAdd(o + 3, v.w * nrm);
}

// ---------------------------------------------------------------------------
// BatchNorm statistics / apply
// ---------------------------------------------------------------------------
__global__ void zero_stats_kernel(float* stats) { stats[threadIdx.x] = 0.0f; }

template <int NC>
__global__ void bn_stats_kernel(const float* __restrict__ agg, float* __restrict__ stats, int n) {
    int gid = blockIdx.x * blockDim.x + threadIdx.x;
    int c = gid % NC;
    int rstart = gid / NC;
    int rstride = (gridDim.x * blockDim.x) / NC;
    float s = 0.0f, ss = 0.0f;
    for (int i = rstart; i < n; i += rstride) {
        float v = agg[(size_t)i * NC + c];
        s += v;
        ss += v * v;
    }
    atomicAdd(&stats[c], s);
    atomicAdd(&stats[NC + c], ss);
}

__global__ void bn_finalize_kernel(float* stats, int nc, float invn) {
    int c = threadIdx.x;
    if (c < nc) {
        float mu = stats[c] * invn;
        float var = stats[nc + c] * invn - mu * mu;
        stats[c] = mu;
        stats[nc + c] = rsqrtf(var + EPS_BN);
    }
}

template <int NC>
__global__ void bn_apply_kernel(float* __restrict__ h, const float* __restrict__ stats,
                                const float* __restrict__ g, const float* __restrict__ be, int n) {
    size_t gid = (size_t)blockIdx.x * blockDim.x + threadIdx.x;
    if (gid >= (size_t)n * NC) return;
    int c = (int)(gid % NC);
    float v = (h[gid] - stats[c]) * stats[NC + c] * g[c] + be[c];
    h[gid] = fmaxf(v, 0.0f);
}

// ---------------------------------------------------------------------------
// Head: bias+ReLU (32 ch), then 32->1 dot per node
// ---------------------------------------------------------------------------
__global__ void relu_bias32_kernel(float* __restrict__ t, const float* __restrict__ b, int n) {
    size_t gid = (size_t)blockIdx.x * blockDim.x + threadIdx.x;
    if (gid >= (size_t)n * 32) return;
    int c = (int)(gid & 31);
    t[gid] = fmaxf(t[gid] + b[c], 0.0f);
}

__global__ void head_dot_kernel(const float* __restrict__ r, const float* __restrict__ wc2,
                                const float* __restrict__ bc2, float* __restrict__ out, int n) {
    int i = blockIdx.x * blockDim.x + threadIdx.x;
    if (i >= n) return;
    const float* row = r + (size_t)i * 32;
    float acc = bc2[0];
#pragma unroll
    for (int c = 0; c < 32; ++c) acc += row[c] * wc2[c];
    out[i] = acc;
}

// ---------------------------------------------------------------------------
// Host-side orchestration
// ---------------------------------------------------------------------------
static inline int cdiv(long long a, long long b) { return (int)((a + b - 1) / b); }

extern "C" void kernel_launch(void* const* d_in, const int* in_sizes, int n_in,
                              void* d_out, int out_size, void* d_ws, size_t ws_size,
                              hipStream_t stream) {
    const float* x = (const float*)d_in[0];
    const long long* ei = (const long long*)d_in[1];
    const float* w0 = (const float*)d_in[2];
    const float* b0 = (const float*)d_in[3];
    const float* g0 = (const float*)d_in[4];
    const float* be0 = (const float*)d_in[5];
    const float* w1 = (const float*)d_in[6];
    const float* b1 = (const float*)d_in[7];
    const float* g1 = (const float*)d_in[8];
    const float* be1 = (const float*)d_in[9];
    const float* w2 = (const float*)d_in[10];
    const float* b2 = (const float*)d_in[11];
    const float* g2 = (const float*)d_in[12];
    const float* be2 = (const float*)d_in[13];
    const float* wc1 = (const float*)d_in[14];
    const float* bc1 = (const float*)d_in[15];
    const float* wc2 = (const float*)d_in[16];
    const float* bc2 = (const float*)d_in[17];
    float* out = (float*)d_out;

    const int N = in_sizes[0] / N_CH0;
    const int E = in_sizes[1] / 2;

    // workspace layout (floats): dis[N] | bufA[N*128] | bufB[N*128] | stats[256]
    float* dis = (float*)d_ws;
    float* bufA = dis + N;
    float* bufB = bufA + (size_t)N * 128;
    float* stats = bufB + (size_t)N * 128;

    const float invn = 1.0f / (float)N;

    // --- degree + symmetric normalization ---
    init_deg_kernel<<<cdiv(N, 256), 256, 0, stream>>>(dis, N);
    deg_scatter_kernel<<<cdiv(E, 256), 256, 0, stream>>>(ei, dis, E);
    finalize_dis_kernel<<<cdiv(N, 256), 256, 0, stream>>>(dis, N);

    // --- layer 0: x(128) -> 128 ---
    gemm_wmma_kernel<128, 128, 4><<<cdiv(N, 64), 256, 0, stream>>>(x, w0, bufA, N);
    agg_init_kernel<128><<<cdiv((long long)N * 128, 256), 256, 0, stream>>>(bufA, dis, b0, bufB, N);
    edge_scatter_kernel<128><<<cdiv((long long)E * 32, 256), 256, 0, stream>>>(ei, bufA, dis, bufB, E);
    zero_stats_kernel<<<1, 256, 0, stream>>>(stats);
    bn_stats_kernel<128><<<512, 256, 0, stream>>>(bufB, stats, N);
    bn_finalize_kernel<<<1, 128, 0, stream>>>(stats, 128, invn);
    bn_apply_kernel<128><<<cdiv((long long)N * 128, 256), 256, 0, stream>>>(bufB, stats, g0, be0, N);

    // --- layer 1: 128 -> 128 ---
    gemm_wmma_kernel<128, 128, 4><<<cdiv(N, 64), 256, 0, stream>>>(bufB, w1, bufA, N);
    agg_init_kernel<128><<<cdiv((long long)N * 128, 256), 256, 0, stream>>>(bufA, dis, b1, bufB, N);
    edge_scatter_kernel<128><<<cdiv((long long)E * 32, 256), 256, 0, stream>>>(ei, bufA, dis, bufB, E);
    zero_stats_kernel<<<1, 256, 0, stream>>>(stats);
    bn_stats_kernel<128><<<512, 256, 0, stream>>>(bufB, stats, N);
    bn_finalize_kernel<<<1, 128, 0, stream>>>(stats, 128, invn);
    bn_apply_kernel<128><<<cdiv((long long)N * 128, 256), 256, 0, stream>>>(bufB, stats, g1, be1, N);

    // --- layer 2: 128 -> 64 ---
    gemm_wmma_kernel<128, 64, 4><<<cdiv(N, 64), 128, 0, stream>>>(bufB, w2, bufA, N);
    agg_init_kernel<64><<<cdiv((long long)N * 64, 256), 256, 0, stream>>>(bufA, dis, b2, bufB, N);
    edge_scatter_kernel<64><<<cdiv((long long)E * 16, 256), 256, 0, stream>>>(ei, bufA, dis, bufB, E);
    zero_stats_kernel<<<1, 256, 0, stream>>>(stats);
    bn_stats_kernel<64><<<512, 256, 0, stream>>>(bufB, stats, N);
    bn_finalize_kernel<<<1, 64, 0, stream>>>(stats, 64, invn);
    bn_apply_kernel<64><<<cdiv((long long)N * 64, 256), 256, 0, stream>>>(bufB, stats, g2, be2, N);

    // --- head: 64 -> 32 (WMMA) -> 1 ---
    gemm_wmma_kernel<64, 32, 4><<<cdiv(N, 64), 64, 0, stream>>>(bufB, wc1, bufA, N);
    relu_bias32_kernel<<<cdiv((long long)N * 32, 256), 256, 0, stream>>>(bufA, bc1, N);
    head_dot_kernel<<<cdiv(N, 256), 256, 0, stream>>>(bufA, wc2, bc2, out, N);
}